// GCN_30227979829928
// MI455X (gfx1250) — compile-verified
//
#include <hip/hip_runtime.h>

typedef __attribute__((ext_vector_type(2))) float v2f;
typedef __attribute__((ext_vector_type(8))) float v8f;

#define N_FEAT 512
#define HIDDEN 128
#define NCLS   64

__device__ __forceinline__ void atomAddF32(float* p, float v) {
    // lowers to global_atomic_add_f32 (no CAS loop)
    unsafeAtomicAdd(p, v);
}

// ---------------- elementwise helpers ----------------

__global__ void fill_kernel(float* __restrict__ p, float v, long long n) {
    long long i = (long long)blockIdx.x * blockDim.x + threadIdx.x;
    if (i < n) p[i] = v;
}

__global__ void deg_accum_kernel(const int* __restrict__ dst, float* __restrict__ deg, int E) {
    int e = blockIdx.x * blockDim.x + threadIdx.x;
    if (e < E) atomAddF32(&deg[dst[e]], 1.0f);
}

__global__ void rsqrt_kernel(float* __restrict__ p, int n) {
    int i = blockIdx.x * blockDim.x + threadIdx.x;
    if (i < n) p[i] = rsqrtf(p[i]);
}

template<int F>
__global__ void bias_relu_kernel(float* __restrict__ t, const float* __restrict__ b, long long n) {
    long long i = (long long)blockIdx.x * blockDim.x + threadIdx.x;
    if (i < n) {
        int j = (int)(i & (F - 1));
        t[i] = fmaxf(t[i] + b[j], 0.0f);
    }
}

template<int F>
__global__ void init_bias_kernel(float* __restrict__ out, const float* __restrict__ b, long long n) {
    long long i = (long long)blockIdx.x * blockDim.x + threadIdx.x;
    if (i < n) out[i] = b[(int)(i & (F - 1))];
}

// ---------------- WMMA f32 GEMM: C[M x NCOL] = A[M x K] @ B[K x NCOL] ----------------
// Register-blocked over N: each wave32 owns one 16-row M-tile and ALL NCOL columns
// (NT = NCOL/16 accumulators). Per k-step: one A fragment load (2 dwords) is reused
// by NT back-to-back v_wmma_f32_16x16x4_f32 ops against NT B fragments. A (the big
// HBM-resident operand) is loaded exactly once per M-tile; B is tiny and L2-resident.
// 8 waves per block = 8 M-tiles per block.

template<int K, int NCOL>
__global__ __launch_bounds__(256)
void gemm_wmma_f32(const float* __restrict__ A, const float* __restrict__ B,
                   float* __restrict__ C, int M) {
    constexpr int NT = NCOL / 16;
    const int lane = threadIdx.x & 31;     // wave32
    const int wave = threadIdx.x >> 5;
    const int m0 = (blockIdx.x * 8 + wave) * 16;
    if (m0 >= M) return;                   // whole-wave exit: EXEC stays all-1s for WMMA

    // A 16x4 frag: lanes 0-15 -> M=lane, K=kb..kb+1 with kb = (lane>>4)*2
    const int mr = lane & 15;
    const int kb = (lane >> 4) << 1;
    // B 4x16 frag: lane holds B[kb][n], B[kb+1][n], n = 16*nt + (lane&15)
    const int nc = lane & 15;

    const float* a_ptr = A + (size_t)(m0 + mr) * K + kb;
    const float* b_ptr = B + (size_t)kb * NCOL + nc;

    v8f acc[NT] = {};
#pragma unroll 4
    for (int k = 0; k < K; k += 4) {
        v2f a;
        a.x = a_ptr[0];
        a.y = a_ptr[1];
#pragma unroll
        for (int nt = 0; nt < NT; ++nt) {
            v2f b;
            b.x = b_ptr[nt * 16];
            b.y = b_ptr[NCOL + nt * 16];
            acc[nt] = __builtin_amdgcn_wmma_f32_16x16x4_f32(
                /*neg_a=*/false, a, /*neg_b=*/false, b,
                /*c_mod=*/(short)0, acc[nt], /*reuse_a=*/false, /*reuse_b=*/false);
        }
        a_ptr += 4;
        b_ptr += (size_t)4 * NCOL;
    }

    // C/D layout: VGPR r -> row m0 + r + (lane>>4)*8, col 16*nt + (lane&15)
    const int rbase = m0 + ((lane >> 4) << 3);
#pragma unroll
    for (int nt = 0; nt < NT; ++nt) {
        float* c_ptr = C + (size_t)rbase * NCOL + nt * 16 + nc;
#pragma unroll
        for (int r = 0; r < 8; ++r)
            c_ptr[(size_t)r * NCOL] = acc[nt][r];
    }
}

// ---------------- edge scatter: acc[dst] += h[src] * dinv[src]*dinv[dst] ----------------
// 2^LOG_TPE threads per edge, each handling 4 contiguous features (float4 gather,
// 4x global_atomic_add_f32). Virtual edges e >= E are the self-loops (src=dst=e-E).

template<int F, int LOG_TPE>
__global__ void scatter_kernel(const float* __restrict__ h, const int* __restrict__ srcI,
                               const int* __restrict__ dstI, const float* __restrict__ dinv,
                               float* __restrict__ acc, int E, int N) {
    long long tid = (long long)blockIdx.x * blockDim.x + threadIdx.x;
    long long edge = tid >> LOG_TPE;
    if (edge >= (long long)E + N) return;
    int fo = ((int)tid & ((1 << LOG_TPE) - 1)) << 2;

    int s, d;
    if (edge < E) { s = srcI[edge]; d = dstI[edge]; }
    else          { s = d = (int)(edge - E); }

    float nrm = dinv[s] * dinv[d];
    const float4 v = *(const float4*)(h + (size_t)s * F + fo);
    float* o = acc + (size_t)d * F + fo;
    atomAddF32(o + 0, v.x * nrm);
    atomAddF32(o + 1, v.y * nrm);
    atomAddF32(o + 2, v.z * nrm);
    atomAddF32(o + 3, v.w * nrm);
}

// ---------------- host launch ----------------

extern "C" void kernel_launch(void* const* d_in, const int* in_sizes, int n_in,
                              void* d_out, int out_size, void* d_ws, size_t ws_size,
                              hipStream_t stream) {
    const float* x  = (const float*)d_in[0];
    const int*   ei = (const int*)d_in[1];
    const float* W1 = (const float*)d_in[2];
    const float* b1 = (const float*)d_in[3];
    const float* W2 = (const float*)d_in[4];
    const float* b2 = (const float*)d_in[5];
    float* out = (float*)d_out;

    const int N = in_sizes[0] / N_FEAT;   // 50000
    const int E = in_sizes[1] / 2;        // 640000
    const int* srcI = ei;
    const int* dstI = ei + E;

    // workspace carve-up (256B aligned): deg | t1 (25.6MB, reused as t3) | t2 (25.6MB)
    char* ws = (char*)d_ws;
    size_t degB = ((size_t)N * sizeof(float) + 255) & ~(size_t)255;
    size_t t1B  = (((size_t)N * HIDDEN * sizeof(float)) + 255) & ~(size_t)255;
    float* deg = (float*)ws;             // becomes dinv in place
    float* t1  = (float*)(ws + degB);
    float* t2  = (float*)(ws + degB + t1B);
    float* t3  = t1;                     // t1 is dead after scatter1

    const int BT = 256;
    const int Mtiles = N / 16;           // 3125 (N divisible by 16)
    const int gemmBlocks = (Mtiles + 7) / 8;

    // 1) deg = 1 (self-loop contribution), then += 1 per incoming edge, then rsqrt
    fill_kernel<<<(N + BT - 1) / BT, BT, 0, stream>>>(deg, 1.0f, N);
    deg_accum_kernel<<<(E + BT - 1) / BT, BT, 0, stream>>>(dstI, deg, E);
    rsqrt_kernel<<<(N + BT - 1) / BT, BT, 0, stream>>>(deg, N);

    // 2) zero aggregation buffer for layer 1
    long long t2n = (long long)N * HIDDEN;
    fill_kernel<<<(int)((t2n + BT - 1) / BT), BT, 0, stream>>>(t2, 0.0f, t2n);

    // 3) GEMM1: t1 = x @ W1   (50000x512 @ 512x128), f32 WMMA, N-register-blocked
    gemm_wmma_f32<N_FEAT, HIDDEN><<<gemmBlocks, 256, 0, stream>>>(x, W1, t1, N);

    // 4) scatter layer 1: t2[dst] += t1[src] * norm  (32 threads/edge)
    long long tot1 = ((long long)E + N) << 5;
    scatter_kernel<HIDDEN, 5><<<(int)((tot1 + BT - 1) / BT), BT, 0, stream>>>(
        t1, srcI, dstI, deg, t2, E, N);

    // 5) h = relu(t2 + b1)
    bias_relu_kernel<HIDDEN><<<(int)((t2n + BT - 1) / BT), BT, 0, stream>>>(t2, b1, t2n);

    // 6) GEMM2: t3 = h @ W2   (50000x128 @ 128x64), f32 WMMA
    gemm_wmma_f32<HIDDEN, NCLS><<<gemmBlocks, 256, 0, stream>>>(t2, W2, t3, N);

    // 7) out rows initialized to b2, then scatter layer 2 (16 threads/edge)
    long long outn = (long long)N * NCLS;
    init_bias_kernel<NCLS><<<(int)((outn + BT - 1) / BT), BT, 0, stream>>>(out, b2, outn);
    long long tot2 = ((long long)E + N) << 4;
    scatter_kernel<NCLS, 4><<<(int)((tot2 + BT - 1) / BT), BT, 0, stream>>>(
        t3, srcI, dstI, deg, out, E, N);
}